// PBATransformerSparseMLP_16569983828105
// MI455X (gfx1250) — compile-verified
//
#include <hip/hip_runtime.h>

#define T_TOTAL (16 * 8192)
#define D_MODEL 128
#define D_HID   512
#define DIN     160
#define BEH     32
#define NEXP    8
#define MT      64            // tokens per block tile
#define NTHREADS 512          // 16 wave32s
#define XS_STRIDE 164         // 160 padded: conflict-free, keeps b128 alignment
#define H_STRIDE  514         // 512 padded: conflict-free, keeps b64 alignment

typedef float v2f __attribute__((ext_vector_type(2)));
typedef float v4f __attribute__((ext_vector_type(4)));
typedef float v8f __attribute__((ext_vector_type(8)));

__global__ void k_zero(int* counts) {
  if (threadIdx.x < NEXP) counts[threadIdx.x] = 0;
}

__global__ void k_build(const int* __restrict__ pos, int* counts, int* lists) {
  int t = blockIdx.x * 256 + threadIdx.x;
  if (t < T_TOTAL) {
    int e = pos[t];
    int slot = atomicAdd(&counts[e], 1);
    lists[e * T_TOTAL + slot] = t;
  }
}

__global__ __launch_bounds__(NTHREADS) void k_moe(
    const float* __restrict__ hidden, const int* __restrict__ behav,
    const float* __restrict__ Wi, const float* __restrict__ Wo,
    const float* __restrict__ emb, const int* __restrict__ counts,
    const int* __restrict__ lists, float* __restrict__ out) {
  // xs (64 x 164 = 42KB, phase 0/1) aliases hs (64 x 514 = 128.5KB, phase 2/3):
  // xs is dead after the mm1 K-loop; a barrier separates last xs read from
  // first hs write, so one buffer serves both.
  __shared__ float smem[MT * H_STRIDE];
  __shared__ int stok[MT];
  __shared__ int sbeh[MT];
  float* xs = smem;
  float* hs = smem;

  const int e    = blockIdx.y;
  const int tile = blockIdx.x;
  const int cnt  = counts[e];
  if (tile * MT >= cnt) return;          // uniform -> EXEC stays full

  const int tid = threadIdx.x;
  if (tid < MT) {
    int gi  = tile * MT + tid;
    int tok = (gi < cnt) ? lists[e * T_TOTAL + gi] : -1;
    stok[tid] = tok;
    sbeh[tid] = (tok >= 0) ? behav[tok] : 0;
  }
  __syncthreads();

  // gather x = concat(hidden, emb[behav]) rows into LDS, 16B granules
  {
    const int NV4 = DIN / 4;             // 40 float4s per row
    for (int idx = tid; idx < MT * NV4; idx += NTHREADS) {
      int row = idx / NV4, col = (idx - row * NV4) * 4;
      int tok = stok[row];
      v4f v = {0.f, 0.f, 0.f, 0.f};
      if (tok >= 0)
        v = (col < D_MODEL)
                ? *(const v4f*)&hidden[(size_t)tok * D_MODEL + col]
                : *(const v4f*)&emb[sbeh[row] * BEH + (col - D_MODEL)];
      *(v4f*)&xs[row * XS_STRIDE + col] = v;
    }
  }
  __syncthreads();

  const int w = tid >> 5, lane = tid & 31, hi = lane >> 4, l15 = lane & 15;
  const v8f vzero = {0.f, 0.f, 0.f, 0.f, 0.f, 0.f, 0.f, 0.f};

  // ---------- mm1: h[64x512] = relu(xs[64x160] * Wi[e]) ----------
  // wave w: all 4 M-tiles x 2 N-tiles {2w, 2w+1}; B frag reused by 4 WMMAs
  const float* wiE = Wi + (size_t)e * DIN * D_HID;
  v8f acc1[4][2];
#pragma unroll
  for (int mt = 0; mt < 4; ++mt)
#pragma unroll
    for (int nt = 0; nt < 2; ++nt) acc1[mt][nt] = vzero;

  for (int k0 = 0; k0 < DIN; k0 += 4) {
    v2f a[4];
#pragma unroll
    for (int mt = 0; mt < 4; ++mt)
      a[mt] = *(const v2f*)&xs[(mt * 16 + l15) * XS_STRIDE + k0 + 2 * hi];
#pragma unroll
    for (int nt = 0; nt < 2; ++nt) {
      int n0 = (2 * w + nt) * 16;
      const float* bp = wiE + (size_t)(k0 + 2 * hi) * D_HID + n0 + l15;
      v2f b; b.x = bp[0]; b.y = bp[D_HID];
#pragma unroll
      for (int mt = 0; mt < 4; ++mt)
        acc1[mt][nt] = __builtin_amdgcn_wmma_f32_16x16x4_f32(
            false, a[mt], false, b, (short)0, acc1[mt][nt], false, false);
    }
  }
  __syncthreads();   // all waves done reading xs; smem may be reused as hs

  // fused ReLU + stage h to LDS
#pragma unroll
  for (int mt = 0; mt < 4; ++mt)
#pragma unroll
    for (int nt = 0; nt < 2; ++nt) {
      int n0 = (2 * w + nt) * 16;
#pragma unroll
      for (int g = 0; g < 8; ++g) {
        float v = acc1[mt][nt][g];
        hs[(mt * 16 + g + 8 * hi) * H_STRIDE + n0 + l15] = v > 0.f ? v : 0.f;
      }
    }
  __syncthreads();

  // ---------- mm2: y[64x128] = h * Wo[e] ----------
  // wave w: N-tile = w&7, M-tiles {2*(w>>3), 2*(w>>3)+1}; B frag reused x2
  const float* woE = Wo + (size_t)e * D_HID * D_MODEL;
  const int nt2 = w & 7;
  const int mtb = (w >> 3) * 2;
  const int n0  = nt2 * 16;
  v8f acc2[2]; acc2[0] = vzero; acc2[1] = vzero;

  for (int k0 = 0; k0 < D_HID; k0 += 4) {
    v2f a[2];
#pragma unroll
    for (int j = 0; j < 2; ++j)
      a[j] = *(const v2f*)&hs[((mtb + j) * 16 + l15) * H_STRIDE + k0 + 2 * hi];
    const float* bp = woE + (size_t)(k0 + 2 * hi) * D_MODEL + n0 + l15;
    v2f b; b.x = bp[0]; b.y = bp[D_MODEL];
#pragma unroll
    for (int j = 0; j < 2; ++j)
      acc2[j] = __builtin_amdgcn_wmma_f32_16x16x4_f32(
          false, a[j], false, b, (short)0, acc2[j], false, false);
  }

  // scatter rows back to out via token list
#pragma unroll
  for (int j = 0; j < 2; ++j) {
#pragma unroll
    for (int g = 0; g < 8; ++g) {
      int row = (mtb + j) * 16 + g + 8 * hi;
      int tok = stok[row];
      if (tok >= 0) out[(size_t)tok * D_MODEL + n0 + l15] = acc2[j][g];
    }
  }
}

extern "C" void kernel_launch(void* const* d_in, const int* in_sizes, int n_in,
                              void* d_out, int out_size, void* d_ws, size_t ws_size,
                              hipStream_t stream) {
  const float* hidden = (const float*)d_in[0];
  const int*   pos    = (const int*)d_in[1];
  const int*   behav  = (const int*)d_in[2];
  const float* Wi     = (const float*)d_in[3];
  const float* Wo     = (const float*)d_in[4];
  const float* emb    = (const float*)d_in[5];
  float* out = (float*)d_out;

  int* counts = (int*)d_ws;                       // 8 ints
  int* lists  = (int*)((char*)d_ws + 256);        // 8 * 131072 ints (4 MB)

  k_zero<<<1, 32, 0, stream>>>(counts);
  k_build<<<(T_TOTAL + 255) / 256, 256, 0, stream>>>(pos, counts, lists);

  dim3 grid(T_TOTAL / MT, NEXP);                  // 2048 tile slots x 8 experts
  k_moe<<<grid, NTHREADS, 0, stream>>>(hidden, behav, Wi, Wo, emb, counts, lists, out);
}